// TemporalShift_49512382988368
// MI455X (gfx1250) — compile-verified
//
#include <hip/hip_runtime.h>

// TemporalShift: out[t,h,d,c] = x[t+1|t-1|t, h,d,c] depending on channel band
// (ch = h*64+c: <96 -> t+1, <192 -> t-1, else copy), zero fill at segment-group
// boundaries (s=t%8). Pure data movement: 154 MB in + 154 MB out -> ~13us floor
// at 23.3 TB/s. Streamed through the CDNA5 async global<->LDS DMA engines
// (ASYNCcnt), with all band/shift logic hoisted out of the transfer loops:
// one block owns one contiguous (t,h) slice of 3136 b128 elements, and with
// blockDim=224 (== 0 mod 16) each thread's channel band is loop-invariant.

typedef __attribute__((ext_vector_type(4))) int v4i;

#define HAS_ASYNC_LDS ( __has_builtin(__builtin_amdgcn_global_load_async_to_lds_b128) \
                     && __has_builtin(__builtin_amdgcn_global_store_async_from_lds_b128) \
                     && __has_builtin(__builtin_amdgcn_s_wait_asynccnt) )

namespace {
constexpr int NT     = 256;
constexpr int HEADS  = 12;
constexpr int DD     = 196;
constexpr int C4     = 16;                  // 64 floats = 16 b128 per (d) row
constexpr int ELEMS  = DD * C4;             // 3136 b128 per (t,h) slice
constexpr int SHIFT4 = HEADS * ELEMS;       // 37,632 b128 per t-step
constexpr int TPB    = 224;                 // 7 waves; 224 % 16 == 0
constexpr int ITERS  = ELEMS / TPB;         // 14 b128 per thread
static_assert(ELEMS % TPB == 0, "slice must tile exactly");
static_assert(TPB % 16 == 0, "c4 must be loop-invariant per thread");
}

__global__ __launch_bounds__(TPB) void temporal_shift_kernel(
    const v4i* __restrict__ x, v4i* __restrict__ out) {
  const unsigned bid = blockIdx.x;             // t*12 + h  (uniform, SGPR)
  const unsigned t   = bid / (unsigned)HEADS;
  const unsigned h   = bid - t * (unsigned)HEADS;
  const unsigned s   = t & 7u;                 // segment within group of 8
  const unsigned tid = threadIdx.x;
  const unsigned ch  = h * 64u + (tid & 15u) * 4u;  // per-thread, loop-invariant

  int  delta;
  bool valid;
  if (ch < 96u)       { delta =  SHIFT4; valid = (s != 7u); }  // shift left
  else if (ch < 192u) { delta = -SHIFT4; valid = (s != 0u); }  // shift right
  else                { delta = 0;       valid = true;      }  // passthrough

  const unsigned base = bid * (unsigned)ELEMS;
  const v4i* src = x + (int)(base + tid) + delta;
  v4i*       dst = out + base + tid;

  __shared__ v4i stage[ELEMS];                 // 49 KB staging slice

#if HAS_ASYNC_LDS
  if (valid) {
    // 14 async DMA loads global -> LDS, constant offsets fold into IOFFSET
#pragma unroll
    for (int j = 0; j < ITERS; ++j)
      __builtin_amdgcn_global_load_async_to_lds_b128(
          (v4i*)(src + j * TPB), &stage[tid + j * TPB], 0, 0);
  } else {
    // boundary zero-fill lanes populate their LDS slots (disjoint EXEC mask)
    v4i z = {0, 0, 0, 0};
#pragma unroll
    for (int j = 0; j < ITERS; ++j)
      stage[tid + j * TPB] = z;
  }
  __builtin_amdgcn_s_wait_asynccnt(0);         // all DMA loads landed in LDS
#if __has_builtin(__builtin_amdgcn_s_wait_dscnt)
  __builtin_amdgcn_s_wait_dscnt(0);            // zero-fill DS stores landed
#endif
  // 14 async DMA stores LDS -> global (all lanes)
#pragma unroll
  for (int j = 0; j < ITERS; ++j)
    __builtin_amdgcn_global_store_async_from_lds_b128(
        dst + j * TPB, &stage[tid + j * TPB], 0, 0);
  // S_ENDPGM performs an implicit wait-idle, draining ASYNCcnt.
#else
  // Fallback: direct nontemporal b128 streaming copy.
  v4i z = {0, 0, 0, 0};
#pragma unroll
  for (int j = 0; j < ITERS; ++j) {
    v4i v = valid ? __builtin_nontemporal_load(src + j * TPB) : z;
    __builtin_nontemporal_store(v, dst + j * TPB);
  }
#endif
}

extern "C" void kernel_launch(void* const* d_in, const int* in_sizes, int n_in,
                              void* d_out, int out_size, void* d_ws, size_t ws_size,
                              hipStream_t stream) {
  (void)in_sizes; (void)n_in; (void)d_ws; (void)ws_size; (void)out_size;
  const v4i* x = (const v4i*)d_in[0];
  v4i* out     = (v4i*)d_out;
  temporal_shift_kernel<<<dim3(NT * HEADS), dim3(TPB), 0, stream>>>(x, out);
}